// SingleNodeReadout_44968307589152
// MI455X (gfx1250) — compile-verified
//
#include <hip/hip_runtime.h>

typedef __attribute__((ext_vector_type(2))) float v2f;
typedef __attribute__((ext_vector_type(8))) float v8f;

#define N_NODES   200000
#define N_PATCHES 256
#define TS        12
#define NHID      16
#define HORIZON   12
#define IN_DIM    204          // 192 + 12
#define IN_PAD    208          // 13 * 16
#define KPATCH    192

// workspace layout (float offsets)
#define PC_OFF    0
#define PC_SZ     (N_PATCHES * IN_PAD)   // 53248
// W1 node-rows, pair-packed: [pair(8)][col(208)][2]  (rows 192..203 of W1, rest 0)
#define W1PP_OFF  (PC_OFF + PC_SZ)
#define W1PP_SZ   (16 * IN_PAD)          // 3328
// W2 pair-packed: [pair(104)][o(16)][2]  (204x12 zero-padded to 208x16)
#define W2PP_OFF  (W1PP_OFF + W1PP_SZ)
#define W2PP_SZ   (IN_PAD * 16)          // 3328
#define B2P_OFF   (W2PP_OFF + W2PP_SZ)
#define PREP_TOT  (W1PP_SZ + W2PP_SZ + 16)

// ---------------------------------------------------------------------------
// Pack weights into WMMA-B-fragment-friendly layouts: K-row pairs interleaved
// so each 4x16 B fragment is one aligned float2 load per lane.
// ---------------------------------------------------------------------------
__global__ void prep_kernel(const float* __restrict__ W1,
                            const float* __restrict__ W2,
                            const float* __restrict__ b2,
                            float* __restrict__ ws) {
    int id = blockIdx.x * blockDim.x + threadIdx.x;
    if (id < W1PP_SZ) {
        int k = id / IN_PAD, j = id % IN_PAD;          // k: 0..15, j: 0..207
        float v = (k < TS && j < IN_DIM) ? W1[(KPATCH + k) * IN_DIM + j] : 0.0f;
        ws[W1PP_OFF + ((k >> 1) * IN_PAD + j) * 2 + (k & 1)] = v;
    } else if (id < W1PP_SZ + W2PP_SZ) {
        int t = id - W1PP_SZ;
        int k = t / 16, o = t % 16;                    // k: 0..207, o: 0..15
        float v = (k < IN_DIM && o < HORIZON) ? W2[k * HORIZON + o] : 0.0f;
        ws[W2PP_OFF + ((k >> 1) * 16 + o) * 2 + (k & 1)] = v;
    } else if (id < PREP_TOT) {
        int o = id - (W1PP_SZ + W2PP_SZ);
        ws[B2P_OFF + o] = (o < HORIZON) ? b2[o] : 0.0f;
    }
}

// ---------------------------------------------------------------------------
// PC[p][j] = b1[j] + sum_{i<192} patch_feat[p][i] * W1[i][j]   (256x208, pad 0)
// patch_feat[p][t*16+f] = mixer_x[((t*256)+p)*16 + f]
// ---------------------------------------------------------------------------
__global__ void pc_kernel(const float* __restrict__ mixer_x,
                          const float* __restrict__ W1,
                          const float* __restrict__ b1,
                          float* __restrict__ ws) {
    __shared__ float pf[KPATCH];
    int p = blockIdx.x;
    int j = threadIdx.x;               // 0..255
    if (j < KPATCH) {
        int t = j >> 4, f = j & 15;
        pf[j] = mixer_x[(t * N_PATCHES + p) * NHID + f];
    }
    __syncthreads();
    float acc = 0.0f;
    if (j < IN_DIM) {
        acc = b1[j];
        for (int i = 0; i < KPATCH; ++i)
            acc = fmaf(pf[i], W1[i * IN_DIM + j], acc);
    }
    if (j < IN_PAD)
        ws[PC_OFF + p * IN_PAD + j] = acc;
}

// ---------------------------------------------------------------------------
// Main readout: one wave32 handles 16 nodes via V_WMMA_F32_16X16X4_F32.
//   phase 1: h(16x208) = relu( PC[patch rows] + X(16x12) @ W1n(12x208) )
//            PC gather loaded straight into the WMMA C operand (D layout).
//   phase 2: out(16x16) = h @ W2p(208x16) + b2p, via LDS re-layout of h.
// 4 waves / block, 64 nodes / block, 3125 blocks.
// ---------------------------------------------------------------------------
__global__ __launch_bounds__(128) void readout_kernel(
        const float* __restrict__ features,
        const int*   __restrict__ node_patch,
        const float* __restrict__ ws,
        float*       __restrict__ out) {
    __shared__ float hbuf[4][16 * IN_PAD];      // 53248 B

    const float* PC   = ws + PC_OFF;
    const v2f*   W1pp = (const v2f*)(ws + W1PP_OFF);   // [pair][col]
    const v2f*   W2pp = (const v2f*)(ws + W2PP_OFF);   // [pair][o]
    const float* b2p  = ws + B2P_OFF;

    const int wave = threadIdx.x >> 5;
    const int lane = threadIdx.x & 31;
    const int m0   = (blockIdx.x * 4 + wave) * 16;

    const int lm = lane & 15;       // column / row-in-half index
    const int hi = lane >> 4;       // lane half
    const int kb = hi * 2;          // K sub-offset in A fragments (K0,K1 | K2,K3)
    const int mh = hi * 8;          // M offset in C/D fragments

    // patch ids of the 8 node rows this lane covers in C/D layout (2x b128)
    int pidx[8];
    {
        const int4* pp = (const int4*)(node_patch + m0 + mh);
        int4 p0 = pp[0];
        int4 p1 = pp[1];
        pidx[0] = p0.x; pidx[1] = p0.y; pidx[2] = p0.z; pidx[3] = p0.w;
        pidx[4] = p1.x; pidx[5] = p1.y; pidx[6] = p1.z; pidx[7] = p1.w;
    }

    // A fragments: node features X (16 rows x K=12), 3 k-chunks of 4 (b64 loads)
    v2f af[3];
    {
        const int row = m0 + lm;
#pragma unroll
        for (int c = 0; c < 3; ++c) {
            const float* fp = features + row * TS + 4 * c + kb;
            af[c].x = fp[0];
            af[c].y = fp[1];
        }
    }

    float* hrow = hbuf[wave];

    // ---------------- phase 1 ----------------
#pragma unroll
    for (int j = 0; j < 13; ++j) {
        const int col = j * 16 + lm;
        v8f acc;
#pragma unroll
        for (int r = 0; r < 8; ++r)                       // C := gathered PC (has b1)
            acc[r] = PC[pidx[r] * IN_PAD + col];
#pragma unroll
        for (int c = 0; c < 3; ++c) {
            v2f bf = W1pp[(2 * c + hi) * IN_PAD + col];   // rows (4c+kb, 4c+kb+1)
            acc = __builtin_amdgcn_wmma_f32_16x16x4_f32(false, af[c], false, bf,
                                                        (short)0, acc, false, false);
        }
#pragma unroll
        for (int r = 0; r < 8; ++r) {                     // relu + stage to LDS
            float v = acc[r];
            hrow[(r + mh) * IN_PAD + col] = v > 0.0f ? v : 0.0f;
        }
    }

    __syncthreads();

    // ---------------- phase 2 ----------------
    v8f oacc;
    {
        const float bb = b2p[lm];
#pragma unroll
        for (int r = 0; r < 8; ++r) oacc[r] = bb;
    }
#pragma unroll 4
    for (int c = 0; c < 52; ++c) {
        const int krow = 4 * c + kb;
        v2f a;
        const float* hp = hrow + lm * IN_PAD + krow;      // h in A layout (ds b64)
        a.x = hp[0];
        a.y = hp[1];
        v2f bf = W2pp[(2 * c + hi) * 16 + lm];            // rows (krow, krow+1)
        oacc = __builtin_amdgcn_wmma_f32_16x16x4_f32(false, a, false, bf,
                                                     (short)0, oacc, false, false);
    }

    if (lm < HORIZON) {
#pragma unroll
        for (int r = 0; r < 8; ++r)
            out[(m0 + r + mh) * HORIZON + lm] = oacc[r];
    }
}

// ---------------------------------------------------------------------------
extern "C" void kernel_launch(void* const* d_in, const int* in_sizes, int n_in,
                              void* d_out, int out_size, void* d_ws, size_t ws_size,
                              hipStream_t stream) {
    const float* mixer_x    = (const float*)d_in[0];
    const float* features   = (const float*)d_in[1];
    const float* W1         = (const float*)d_in[2];
    const float* b1         = (const float*)d_in[3];
    const float* W2         = (const float*)d_in[4];
    const float* b2         = (const float*)d_in[5];
    const int*   node_patch = (const int*)d_in[6];
    float* ws   = (float*)d_ws;
    float* outp = (float*)d_out;

    prep_kernel<<<(PREP_TOT + 255) / 256, 256, 0, stream>>>(W1, W2, b2, ws);
    pc_kernel<<<N_PATCHES, 256, 0, stream>>>(mixer_x, W1, b1, ws);
    readout_kernel<<<N_NODES / 64, 128, 0, stream>>>(features, node_patch, ws, outp);
}